// EXPNN_48498770707161
// MI455X (gfx1250) — compile-verified
//
#include <hip/hip_runtime.h>

typedef float v2f __attribute__((ext_vector_type(2)));
typedef float v8f __attribute__((ext_vector_type(8)));

#define DDIM 128

// ---------------------------------------------------------------- utilities
__global__ void EXPNN_zero_f32(float* __restrict__ p, int n) {
    int i = blockIdx.x * blockDim.x + threadIdx.x;
    if (i < n) p[i] = 0.0f;
}

// ---------------------------------------------------------------- degrees
__global__ void EXPNN_degree(const int* __restrict__ node_idx,
                             const int* __restrict__ hedge_idx,
                             float* __restrict__ Dn, float* __restrict__ Be,
                             int n_inc) {
    int i = blockIdx.x * blockDim.x + threadIdx.x;
    if (i < n_inc) {
        atomicAdd(&Dn[node_idx[i]], 1.0f);
        atomicAdd(&Be[hedge_idx[i]], 1.0f);
    }
}

__global__ void EXPNN_invert(float* __restrict__ p, int n) {
    int i = blockIdx.x * blockDim.x + threadIdx.x;
    if (i < n) {
        float v = p[i];
        p[i] = (v > 0.0f) ? (1.0f / v) : 0.0f;
    }
}

// ---------------------------------------------------------------- WMMA GEMM
// OUT[M x 128] = X[M x 128] * W[128 x 128], f32 accum via V_WMMA_F32_16X16X4_F32.
// One wave per 16-row M-tile; 8 N-tiles (full 128 cols) held in 8 v8f accums.
__global__ void EXPNN_gemm_wmma(const float* __restrict__ X,
                                const float* __restrict__ W,
                                float* __restrict__ OUT, int mtiles) {
    int wave  = threadIdx.x >> 5;           // blockDim.x == 128 -> 4 waves
    int lane  = threadIdx.x & 31;
    int mTile = blockIdx.x * 4 + wave;
    if (mTile >= mtiles) return;            // uniform per wave: EXEC stays full

    int row0 = mTile * 16;
    int half = lane >> 4;                   // 0: lanes 0-15, 1: lanes 16-31
    int lm   = lane & 15;

    const float* xr = X + (size_t)(row0 + lm) * DDIM;

    v8f zero8 = {0.f, 0.f, 0.f, 0.f, 0.f, 0.f, 0.f, 0.f};
    v8f acc[8];
#pragma unroll
    for (int n = 0; n < 8; ++n) acc[n] = zero8;

    for (int k0 = 0; k0 < DDIM; k0 += 4) {
        // A fragment (16x4 f32): lane holds row row0+lm, cols k0+2*half, +1
        v2f a;
        a.x = xr[k0 + 2 * half];
        a.y = xr[k0 + 2 * half + 1];
        const float* wr = W + (size_t)(k0 + 2 * half) * DDIM;
#pragma unroll
        for (int n = 0; n < 8; ++n) {
            // B fragment (4x16 f32): lane holds col n*16+lm, rows k0+2*half, +1
            v2f b;
            b.x = wr[n * 16 + lm];
            b.y = wr[DDIM + n * 16 + lm];
            acc[n] = __builtin_amdgcn_wmma_f32_16x16x4_f32(
                false, a, false, b, (short)0, acc[n], false, false);
        }
    }

    // D layout: VGPR r -> row = row0 + r + 8*half, col = n*16 + lm
#pragma unroll
    for (int n = 0; n < 8; ++n) {
#pragma unroll
        for (int r = 0; r < 8; ++r) {
            OUT[(size_t)(row0 + r + 8 * half) * DDIM + n * 16 + lm] = acc[n][r];
        }
    }
}

// ---------------------------------------------------------------- scatter
// Wave per incidence: gather one 128-float row (float4/lane), optional scale
// by scale[src_segment], scatter-add into dst row via f32 atomics (L2-resident).
__global__ void EXPNN_scatter_rows(const float* __restrict__ src,
                                   const int* __restrict__ seg_src,
                                   const int* __restrict__ seg_dst,
                                   const float* __restrict__ scale,
                                   float* __restrict__ dst, int n_inc) {
    int gid  = blockIdx.x * blockDim.x + threadIdx.x;
    int i    = gid >> 5;
    int lane = gid & 31;
    if (i >= n_inc) return;                 // uniform per wave

    int s = seg_src[i];
    int d = seg_dst[i];
    const float4 v = *(const float4*)(src + (size_t)s * DDIM + lane * 4);
    float m = scale ? scale[s] : 1.0f;
    float* dp = dst + (size_t)d * DDIM + lane * 4;
    atomicAdd(dp + 0, v.x * m);
    atomicAdd(dp + 1, v.y * m);
    atomicAdd(dp + 2, v.z * m);
    atomicAdd(dp + 3, v.w * m);
}

// ---------------------------------------------------------------- finalize
__global__ void EXPNN_finalize_relu(float* __restrict__ h,
                                    const float* __restrict__ Dinv,
                                    const float* __restrict__ bias, int n) {
    int i = blockIdx.x * blockDim.x + threadIdx.x;
    if (i < n) {
        int r = i >> 7, c = i & 127;
        float v = h[i] * Dinv[r] + bias[c];
        h[i] = (v > 0.0f) ? v : 0.0f;
    }
}

__global__ void EXPNN_finalize_relu_mean(const float* __restrict__ h,
                                         const float* __restrict__ Dinv,
                                         const float* __restrict__ bias,
                                         float* __restrict__ out,
                                         int n_rows, float invN) {
    int tid = blockIdx.x * blockDim.x + threadIdx.x;
    int c = tid & 127;
    int r0 = tid >> 7;
    int rstride = (gridDim.x * blockDim.x) >> 7;
    float b = bias[c];
    float sum = 0.0f;
    for (int r = r0; r < n_rows; r += rstride) {
        float v = h[(size_t)r * DDIM + c] * Dinv[r] + b;
        sum += (v > 0.0f) ? v : 0.0f;
    }
    atomicAdd(&out[c], sum * invN);
}

// ---------------------------------------------------------------- launch
extern "C" void kernel_launch(void* const* d_in, const int* in_sizes, int n_in,
                              void* d_out, int out_size, void* d_ws, size_t ws_size,
                              hipStream_t stream) {
    const float* x  = (const float*)d_in[0];
    const int*   ei = (const int*)d_in[1];
    const float* w1 = (const float*)d_in[2];
    const float* b1 = (const float*)d_in[3];
    const float* w2 = (const float*)d_in[4];
    const float* b2 = (const float*)d_in[5];
    float* out = (float*)d_out;

    int n_inc   = in_sizes[1] / 2;
    int n_nodes = in_sizes[0] / DDIM;       // hyperedges share the same count
    const int* node_idx  = ei;              // edge_index[0]
    const int* hedge_idx = ei + n_inc;      // edge_index[1]

    size_t NE = (size_t)n_nodes * DDIM;
    float* ws   = (float*)d_ws;
    float* xw   = ws;                       // [n_nodes, 128]
    float* e    = ws + NE;                  // [n_hedges, 128]
    float* h    = ws + 2 * NE;              // [n_nodes, 128]
    float* Dinv = ws + 3 * NE;              // [n_nodes]  (deg -> inverse in place)
    // Binv = Dinv + n_nodes

    int nelem = (int)NE;
    int eb  = (nelem + 255) / 256;
    int db  = (2 * n_nodes + 255) / 256;
    int ib  = (n_inc + 255) / 256;
    int mtiles = (n_nodes + 15) / 16;
    int gb  = (mtiles + 3) / 4;
    int sb  = (int)(((size_t)n_inc * 32 + 255) / 256);

    // degrees -> inverse degrees (shared by both layers)
    EXPNN_zero_f32<<<db, 256, 0, stream>>>(Dinv, 2 * n_nodes);
    EXPNN_degree<<<ib, 256, 0, stream>>>(node_idx, hedge_idx, Dinv, Dinv + n_nodes, n_inc);
    EXPNN_invert<<<db, 256, 0, stream>>>(Dinv, 2 * n_nodes);
    float* Binv = Dinv + n_nodes;

    // ---- layer 1: h = relu(D^-1 H B^-1 H^T (x W1) + b1)
    EXPNN_gemm_wmma<<<gb, 128, 0, stream>>>(x, w1, xw, mtiles);
    EXPNN_zero_f32<<<eb, 256, 0, stream>>>(e, nelem);
    EXPNN_scatter_rows<<<sb, 256, 0, stream>>>(xw, node_idx, hedge_idx, nullptr, e, n_inc);
    EXPNN_zero_f32<<<eb, 256, 0, stream>>>(h, nelem);
    EXPNN_scatter_rows<<<sb, 256, 0, stream>>>(e, hedge_idx, node_idx, Binv, h, n_inc);
    EXPNN_finalize_relu<<<eb, 256, 0, stream>>>(h, Dinv, b1, nelem);

    // ---- layer 2 + mean over nodes
    EXPNN_gemm_wmma<<<gb, 128, 0, stream>>>(h, w2, xw, mtiles);
    EXPNN_zero_f32<<<eb, 256, 0, stream>>>(e, nelem);
    EXPNN_scatter_rows<<<sb, 256, 0, stream>>>(xw, node_idx, hedge_idx, nullptr, e, n_inc);
    EXPNN_zero_f32<<<eb, 256, 0, stream>>>(h, nelem);
    EXPNN_scatter_rows<<<sb, 256, 0, stream>>>(e, hedge_idx, node_idx, Binv, h, n_inc);
    EXPNN_zero_f32<<<1, 128, 0, stream>>>(out, 128);
    EXPNN_finalize_relu_mean<<<1024, 256, 0, stream>>>(h, Dinv, b2, out, n_nodes,
                                                       1.0f / (float)n_nodes);
}